// SCELoss_57543971832312
// MI455X (gfx1250) — compile-verified
//
#include <hip/hip_runtime.h>
#include <stdint.h>

// ---- problem constants (fixed by the reference) ----
#define NCLS        19
#define HW          (512 * 1024)            // per-image channel plane = 524288 floats
#define NIMG        8
#define TOTAL_PIX   ((long long)NIMG * HW)  // 4,194,304
#define TILE        512                     // pixels per block
#define THREADS     256                     // 8 wave32s
#define IGNORE_LBL  255
#define NEG_LOG_OH_MIN 9.210340371976184f   // -log(1e-4)
#define P_MIN_F     1e-7f

typedef uint32_t u32x4 __attribute__((ext_vector_type(4)));
typedef uint32_t u32x8 __attribute__((ext_vector_type(8)));

// acc[0] = sum(-logp_y), acc[1] = sum(rce), acc[2] = n_valid
__global__ void sce_zero(double* acc) {
    if (threadIdx.x < 3) acc[threadIdx.x] = 0.0;
}

__global__ __launch_bounds__(THREADS)
void sce_main(const float* __restrict__ pred,
              const long long* __restrict__ labels,
              double* __restrict__ acc)
{
    __shared__ float tile[NCLS * TILE];          // 38 KB: 19 channel rows x 512 px
    __shared__ float red_ce [THREADS / 32];
    __shared__ float red_rce[THREADS / 32];
    __shared__ float red_cnt[THREADS / 32];

    const int t = threadIdx.x;
    const long long p0 = (long long)blockIdx.x * TILE;   // first flat pixel of tile
    const int       n  = (int)(p0 / HW);                 // image index (tile never crosses)
    const long long hw0 = p0 % HW;

    // ---- CDNA5 Tensor Data Mover: one 2-D D# per block, issued by wave 0 ----
    // Tile = 19 rows (channels) x 512 floats, row stride = HW floats.
    if (t < 32) {
        const float* gbase =
            pred + ((size_t)n * NCLS) * (size_t)HW + (size_t)hw0;  // tile start
        const uint64_t gaddr    = (uint64_t)(uintptr_t)gbase;      // byte address
        const uint32_t lds_addr = (uint32_t)(uintptr_t)&tile[0];   // LDS byte offset

        // D# group 0 (128b): count=1 | lds_addr | global_addr[56:0] | type=2
        u32x4 g0;
        g0[0] = 1u;                                   // count=1, is_restore=0, gather=0
        g0[1] = lds_addr;                             // bits [63:32]  lds_addr
        g0[2] = (uint32_t)gaddr;                      // bits [95:64]  global_addr lo
        g0[3] = ((uint32_t)(gaddr >> 32) & 0x01FFFFFFu) // bits [120:96] global_addr hi
              | (2u << 30);                           // bits [127:126] type=2 (image)

        // D# group 1 (256b)
        u32x8 g1;
        g1[0] = (2u << 16);                           // wg_mask=0 (not in cluster), data_size=4B
        g1[1] = (uint32_t)((HW & 0xFFFFu) << 16);     // tensor_dim0[15:0]  @ bits[63:48]
        g1[2] = (uint32_t)((HW >> 16) & 0xFFFFu)      // tensor_dim0[31:16] @ bits[79:64]
              | ((uint32_t)NCLS << 16);               // tensor_dim1[15:0]  @ bits[95:80]
        g1[3] = ((uint32_t)TILE << 16);               // tile_dim0=512      @ bits[127:112]
        g1[4] = (uint32_t)NCLS;                       // tile_dim1=19       @ bits[143:128]
        g1[5] = (uint32_t)HW;                         // tensor_dim0_stride lo @ bits[191:160]
        g1[6] = 0u;                                   // stride hi / dim1_stride lo
        g1[7] = 0u;                                   // dim1_stride hi (unused, 2-D)

        asm volatile("tensor_load_to_lds %0, %1" :: "s"(g0), "s"(g1) : "memory");
        __builtin_amdgcn_s_wait_tensorcnt(0);
    }

    // labels for this thread's two pixels overlap the TDM transfer
    const longlong2 lv = ((const longlong2*)labels)[(p0 >> 1) + t];

    __syncthreads();   // publish TDM-written LDS to all 8 waves

    float ce_acc = 0.f, rce_acc = 0.f, cnt_acc = 0.f;
#pragma unroll
    for (int q = 0; q < 2; ++q) {
        const int pix = 2 * t + q;
        const long long lab = (q == 0) ? lv.x : lv.y;
        const bool valid = (lab != IGNORE_LBL);
        const int  y     = valid ? (int)lab : 0;

        float x[NCLS];
#pragma unroll
        for (int c = 0; c < NCLS; ++c) x[c] = tile[c * TILE + pix];

        float m = x[0];
#pragma unroll
        for (int c = 1; c < NCLS; ++c) m = fmaxf(m, x[c]);

        float s = 0.f;
        float xy = x[0];
#pragma unroll
        for (int c = 0; c < NCLS; ++c) {
            s += __expf(x[c] - m);
            if (c == y) xy = x[c];        // unrolled cndmask chain, no scratch
        }
        const float lse   = m + __logf(s);
        const float logpy = xy - lse;

        float psum = 0.f;
#pragma unroll
        for (int c = 0; c < NCLS; ++c) {
            float pc = __expf(x[c] - lse);
            psum += fmaxf(fminf(pc, 1.0f), P_MIN_F);
        }
        const float py = fmaxf(fminf(__expf(logpy), 1.0f), P_MIN_F);

        if (valid) {
            ce_acc  += -logpy;
            rce_acc += NEG_LOG_OH_MIN * (psum - py);
            cnt_acc += 1.0f;
        }
    }

    // ---- wave32 tree reduction ----
#pragma unroll
    for (int off = 16; off >= 1; off >>= 1) {
        ce_acc  += __shfl_down(ce_acc,  off, 32);
        rce_acc += __shfl_down(rce_acc, off, 32);
        cnt_acc += __shfl_down(cnt_acc, off, 32);
    }
    const int wave = t >> 5;
    if ((t & 31) == 0) {
        red_ce [wave] = ce_acc;
        red_rce[wave] = rce_acc;
        red_cnt[wave] = cnt_acc;
    }
    __syncthreads();
    if (t == 0) {
        float ce = 0.f, rc = 0.f, cn = 0.f;
#pragma unroll
        for (int w = 0; w < THREADS / 32; ++w) {
            ce += red_ce[w]; rc += red_rce[w]; cn += red_cnt[w];
        }
        atomicAdd(&acc[0], (double)ce);
        atomicAdd(&acc[1], (double)rc);
        atomicAdd(&acc[2], (double)cn);
    }
}

__global__ void sce_final(const double* __restrict__ acc, float* __restrict__ out) {
    const double nv = acc[2];
    out[0] = (float)(0.1 * acc[0] / nv + 1.0 * acc[1] / nv);
}

extern "C" void kernel_launch(void* const* d_in, const int* in_sizes, int n_in,
                              void* d_out, int out_size, void* d_ws, size_t ws_size,
                              hipStream_t stream) {
    (void)in_sizes; (void)n_in; (void)out_size; (void)ws_size;
    const float*      pred   = (const float*)d_in[0];
    const long long*  labels = (const long long*)d_in[1];
    double*           acc    = (double*)d_ws;

    sce_zero<<<1, 32, 0, stream>>>(acc);
    const int nblocks = (int)(TOTAL_PIX / TILE);   // 8192
    sce_main<<<nblocks, THREADS, 0, stream>>>(pred, labels, acc);
    sce_final<<<1, 1, 0, stream>>>(acc, (float*)d_out);
}